// Color_Loss_29815662969477
// MI455X (gfx1250) — compile-verified
//
#include <hip/hip_runtime.h>

typedef __attribute__((ext_vector_type(2))) float v2f;
typedef __attribute__((ext_vector_type(4))) float v4f;
typedef __attribute__((ext_vector_type(8))) float v8f;

#define NPIX   (32u * 512u * 512u)   // 8,388,608 pixels per channel
#define PLANE  (512u * 512u)         // 262,144 (channel stride)
#define IMG    (3u * PLANE)          // 786,432 (image stride)
#define N4     (NPIX / 4u)           // 2,097,152 float4 groups
#define BLOCKS 1024
#define TPB    256
#define WAVES  (TPB / 32)

// Wave32 all-lane reduction using V_WMMA_F32_16X16X4_F32 as a folding engine.
// A = ones(16x4), B = {v, 0} per lane  =>  every row of D = column sums of B.
// Column n of B is fed exactly by lanes {n, n+16} (row-striped layout), so
// D-vgpr0 in lane L holds v[L%16] + v[L%16+16]. Four xor-folds finish 16 -> 1.
// Requires EXEC == all ones (call only from fully-active, non-divergent code).
__device__ __forceinline__ float wave_reduce_all(float v) {
    v2f a; a[0] = 1.0f; a[1] = 1.0f;
    v2f b; b[0] = v;    b[1] = 0.0f;
    v8f c = {};
    v8f d = __builtin_amdgcn_wmma_f32_16x16x4_f32(
        /*neg_a=*/false, a, /*neg_b=*/false, b,
        /*c_mod=*/(short)0, c, /*reuse_a=*/false, /*reuse_b=*/false);
    float f = d[0];                 // half-wave fold done by the matrix unit
    f += __shfl_xor(f, 1);
    f += __shfl_xor(f, 2);
    f += __shfl_xor(f, 4);
    f += __shfl_xor(f, 8);
    return f;                       // total over all 32 lanes, in every lane
}

__device__ __forceinline__ float smooth_l1(float x) {
    float ax = fabsf(x);
    return (ax < 1.0f) ? 0.5f * ax * ax : ax - 0.5f;
}

__global__ void __launch_bounds__(TPB)
color_loss_partial(const float* __restrict__ real,
                   const float* __restrict__ fake,
                   float* __restrict__ partial) {
    const unsigned tid      = blockIdx.x * TPB + threadIdx.x;
    const unsigned nthreads = gridDim.x * TPB;

    float sy = 0.0f, su = 0.0f, sv = 0.0f;

    for (unsigned g = tid; g < N4; g += nthreads) {
        const unsigned p    = g << 2;          // first pixel of the group
        const unsigned bimg = p >> 18;         // p / PLANE
        const unsigned hw   = p & (PLANE - 1u);
        const unsigned idx  = bimg * IMG + hw; // same image-plane for all 4 pixels

        // Streaming, read-once data: non-temporal 128-bit loads (TH=NT).
        const v4f rR = __builtin_nontemporal_load((const v4f*)(real + idx));
        const v4f rG = __builtin_nontemporal_load((const v4f*)(real + idx + PLANE));
        const v4f rB = __builtin_nontemporal_load((const v4f*)(real + idx + 2u * PLANE));
        const v4f fR = __builtin_nontemporal_load((const v4f*)(fake + idx));
        const v4f fG = __builtin_nontemporal_load((const v4f*)(fake + idx + PLANE));
        const v4f fB = __builtin_nontemporal_load((const v4f*)(fake + idx + 2u * PLANE));

#pragma unroll
        for (int k = 0; k < 4; ++k) {
            const float dR = rR[k] - fR[k];
            const float dG = rG[k] - fG[k];
            const float dB = rB[k] - fB[k];
            // Linear map: yuv(real)-yuv(fake) = M*(real-fake)
            const float dY = fmaf(0.299f,       dR, fmaf( 0.587f,      dG,  0.114f      * dB));
            const float dU = fmaf(-0.14714119f, dR, fmaf(-0.28886916f, dG,  0.43601035f * dB));
            const float dV = fmaf(0.61497657f,  dR, fmaf(-0.51496512f, dG, -0.10001026f * dB));
            sy += fabsf(dY);
            su += smooth_l1(dU);
            sv += smooth_l1(dV);
        }
    }

    // Uniform epilogue: every lane of every wave reaches here (EXEC all ones).
    sy = wave_reduce_all(sy);
    su = wave_reduce_all(su);
    sv = wave_reduce_all(sv);

    __shared__ float smem[WAVES][3];
    const unsigned lane = threadIdx.x & 31u;
    const unsigned wave = threadIdx.x >> 5;
    if (lane == 0) {
        smem[wave][0] = sy;
        smem[wave][1] = su;
        smem[wave][2] = sv;
    }
    __syncthreads();
    if (threadIdx.x == 0) {
        float ty = 0.0f, tu = 0.0f, tv = 0.0f;
#pragma unroll
        for (int w = 0; w < WAVES; ++w) {
            ty += smem[w][0];
            tu += smem[w][1];
            tv += smem[w][2];
        }
        partial[blockIdx.x * 3 + 0] = ty;
        partial[blockIdx.x * 3 + 1] = tu;
        partial[blockIdx.x * 3 + 2] = tv;
    }
}

__global__ void __launch_bounds__(TPB)
color_loss_final(const float* __restrict__ partial,
                 float* __restrict__ out,
                 int nblocks) {
    float sy = 0.0f, su = 0.0f, sv = 0.0f;
    for (int i = (int)threadIdx.x; i < nblocks; i += TPB) {
        sy += partial[3 * i + 0];
        su += partial[3 * i + 1];
        sv += partial[3 * i + 2];
    }

    sy = wave_reduce_all(sy);
    su = wave_reduce_all(su);
    sv = wave_reduce_all(sv);

    __shared__ float smem[WAVES][3];
    const unsigned lane = threadIdx.x & 31u;
    const unsigned wave = threadIdx.x >> 5;
    if (lane == 0) {
        smem[wave][0] = sy;
        smem[wave][1] = su;
        smem[wave][2] = sv;
    }
    __syncthreads();
    if (threadIdx.x == 0) {
        float ty = 0.0f, tu = 0.0f, tv = 0.0f;
#pragma unroll
        for (int w = 0; w < WAVES; ++w) {
            ty += smem[w][0];
            tu += smem[w][1];
            tv += smem[w][2];
        }
        // mean(Y)+mean(U)+mean(V); all three share N = NPIX
        out[0] = (ty + tu + tv) * (1.0f / (float)NPIX);
    }
}

extern "C" void kernel_launch(void* const* d_in, const int* in_sizes, int n_in,
                              void* d_out, int out_size, void* d_ws, size_t ws_size,
                              hipStream_t stream) {
    const float* real = (const float*)d_in[0];
    const float* fake = (const float*)d_in[1];
    float* out     = (float*)d_out;
    float* partial = (float*)d_ws;   // BLOCKS * 3 floats = 12 KB scratch

    color_loss_partial<<<BLOCKS, TPB, 0, stream>>>(real, fake, partial);
    color_loss_final<<<1, TPB, 0, stream>>>(partial, out, BLOCKS);
}